// CenterLoss_13597866459219
// MI455X (gfx1250) — compile-verified
//
#include <hip/hip_runtime.h>
#include <hip/hip_bf16.h>

// CenterLoss collapsed: loss = (1/B) * sum_b clip(||x_b - c_{l_b}||^2) + (C-1)*1e-12
// Bandwidth-bound (~16 MB @ 23.3 TB/s ~ 0.7us). Per-row squared distances are
// computed as diag(E * E^T) on the WMMA f32 16x16x4 path.

typedef __attribute__((ext_vector_type(2))) float v2f;
typedef __attribute__((ext_vector_type(8))) float v8f;

namespace {
constexpr int BATCH   = 4096;
constexpr int FEAT    = 512;
constexpr int CLASSES = 10000;
constexpr int KSLICES = 4;                       // K split per 16-row group
constexpr int KCHUNK  = FEAT / KSLICES;          // 128 floats per wave
constexpr int BLOCK_B = 256;                     // 8 waves / block
constexpr int WAVES   = (BATCH / 16) * KSLICES;  // 1024 waves
constexpr int GRID_B  = WAVES / (BLOCK_B / 32);  // 128 blocks
}

// Each wave handles 16 rows x 128 K-elems. A-matrix f32 16x4 layout (ISA 7.12.2):
// lanes 0-15 hold K=0,1 in vgpr0,1 ; lanes 16-31 hold K=2,3. B (4x16) follows the
// C/D convention (vgpr strides K by 1, lane-half offset = rows/2 = 2), which makes
// the per-lane B values identical to A — one v2f feeds both operands of E*E^T.
__global__ void __launch_bounds__(BLOCK_B)
center_dist_kernel(const float* __restrict__ x,
                   const int*   __restrict__ labels,
                   const float* __restrict__ centers,
                   float*       __restrict__ part)
{
    const int lane     = threadIdx.x & 31;
    const int wid      = blockIdx.x * (blockDim.x >> 5) + (threadIdx.x >> 5);
    const int rowGroup = wid >> 2;          // 16-row group
    const int kslice   = wid & (KSLICES - 1);
    const int r0       = rowGroup << 4;
    const int row      = r0 + (lane & 15);  // lanes 16-31 mirror lanes 0-15
    const int kbase    = kslice * KCHUNK;

    const int lab = labels[row];
    const float4* __restrict__ xr = (const float4*)(x       + (size_t)row * FEAT + kbase);
    const float4* __restrict__ cr = (const float4*)(centers + (size_t)lab * FEAT + kbase);
    const bool hi = lane >= 16;

    v8f acc = {};
#pragma unroll 4
    for (int k = 0; k < KCHUNK / 4; ++k) {
        float4 xv = xr[k];
        float4 cv = cr[k];
        float e0 = xv.x - cv.x;
        float e1 = xv.y - cv.y;
        float e2 = xv.z - cv.z;
        float e3 = xv.w - cv.w;
        v2f ab;
        ab[0] = hi ? e2 : e0;
        ab[1] = hi ? e3 : e1;
        // acc += A(16x4) * B(4x16), A == B == E-chunk  -> diag holds ||e||^2 partials
        acc = __builtin_amdgcn_wmma_f32_16x16x4_f32(
                  false, ab, false, ab, (short)0, acc, false, false);
    }

    // C/D layout: lanes 0-15: vgpr j -> (M=j,   N=lane);
    //             lanes16-31: vgpr j -> (M=j+8, N=lane-16).
    // Diagonal M==N lives in: lane m (m<8) at vgpr m; lane m+16 (m>=8) at vgpr m-8.
    const int j = lane & 7;
    float d = acc[0];
    if (j == 1) d = acc[1];
    if (j == 2) d = acc[2];
    if (j == 3) d = acc[3];
    if (j == 4) d = acc[4];
    if (j == 5) d = acc[5];
    if (j == 6) d = acc[6];
    if (j == 7) d = acc[7];

    const bool valid = (lane < 8) || (lane >= 24);
    const int  orow  = (lane < 8) ? (r0 + lane) : (r0 + lane - 16);
    if (valid)
        part[(size_t)orow * KSLICES + kslice] = d;
}

// Deterministic finalize: fixed-order per-row recombine, clip, fixed-order tree sum.
__global__ void __launch_bounds__(256)
center_reduce_kernel(const float* __restrict__ part, float* __restrict__ out)
{
    __shared__ float red[256];
    const int t = threadIdx.x;
    float s = 0.0f;
    for (int r = t; r < BATCH; r += 256) {
        const float* p = part + (size_t)r * KSLICES;
        float d = (p[0] + p[1]) + (p[2] + p[3]);
        d = fminf(fmaxf(d, 1e-12f), 1e12f);
        s += d;
    }
    red[t] = s;
    __syncthreads();
    for (int off = 128; off > 0; off >>= 1) {
        if (t < off) red[t] += red[t + off];
        __syncthreads();
    }
    if (t == 0)
        out[0] = red[0] * (1.0f / (float)BATCH)
               + (float)((double)(CLASSES - 1) * 1e-12);  // (B*C - B)*1e-12 / B
}

extern "C" void kernel_launch(void* const* d_in, const int* in_sizes, int n_in,
                              void* d_out, int out_size, void* d_ws, size_t ws_size,
                              hipStream_t stream) {
    const float* x       = (const float*)d_in[0];
    const int*   labels  = (const int*)d_in[1];
    const float* centers = (const float*)d_in[2];
    float*       out     = (float*)d_out;
    float*       part    = (float*)d_ws;   // BATCH * KSLICES floats = 64 KB

    center_dist_kernel<<<GRID_B, BLOCK_B, 0, stream>>>(x, labels, centers, part);
    center_reduce_kernel<<<1, 256, 0, stream>>>(part, out);
    (void)in_sizes; (void)n_in; (void)out_size; (void)ws_size;
}